// rnnDecoder_29368986370322
// MI455X (gfx1250) — compile-verified
//
#include <hip/hip_runtime.h>
#include <hip/hip_bf16.h>

typedef __attribute__((ext_vector_type(2))) float v2f;
typedef __attribute__((ext_vector_type(8))) float v8f;
typedef __attribute__((ext_vector_type(4))) unsigned int v4u;
typedef __attribute__((ext_vector_type(8))) unsigned int v8u;

#define STATE 1024
#define EMB   512
#define VOC   32000
#define TT    512
#define G4    4096          // 4*STATE
#define WIH_LD 2560         // EMB + 2*STATE

// ---------------- persistent-grid barrier ----------------
#define SCAN_BLOCKS 128

__device__ __forceinline__ void grid_barrier(unsigned* cnt, unsigned* gen, unsigned nblocks) {
    __threadfence();
    __syncthreads();
    if (threadIdx.x == 0) {
        unsigned g = __hip_atomic_load(gen, __ATOMIC_RELAXED, __HIP_MEMORY_SCOPE_AGENT);
        unsigned a = __hip_atomic_fetch_add(cnt, 1u, __ATOMIC_ACQ_REL, __HIP_MEMORY_SCOPE_AGENT);
        if (a == nblocks - 1u) {
            __hip_atomic_store(cnt, 0u, __ATOMIC_RELAXED, __HIP_MEMORY_SCOPE_AGENT);
            __hip_atomic_fetch_add(gen, 1u, __ATOMIC_RELEASE, __HIP_MEMORY_SCOPE_AGENT);
        } else {
            while (__hip_atomic_load(gen, __ATOMIC_ACQUIRE, __HIP_MEMORY_SCOPE_AGENT) == g)
                __builtin_amdgcn_s_sleep(1);
        }
    }
    __syncthreads();
}

__device__ __forceinline__ float sigmoidf_(float x) { return 1.0f / (1.0f + expf(-x)); }

// ---------------- Tensor Data Mover: 2D fp32 tile -> LDS ----------------
// Loads `rows` x `rowlen` fp32 tile (row pitch `pitch_elems`) into LDS at
// `lds_addr`, adding 4 DWORDs of LDS padding after every row (pad_interval
// code `pint`: rows are 2^(pint+1) DWORDs) so the LDS row stride is
// rowlen+4 floats (bank-conflict-free WMMA feeds).
__device__ __forceinline__ void tdm_load_tile(unsigned lds_addr, const float* gaddr,
                                              unsigned rows, unsigned rowlen,
                                              unsigned pitch_elems, unsigned pint) {
    unsigned long long ga = (unsigned long long)(uintptr_t)gaddr;
    v4u g0;
    g0[0] = 1u;                                              // count=1 (valid user D#)
    g0[1] = lds_addr;                                        // lds_addr [63:32]
    g0[2] = (unsigned)(ga & 0xFFFFFFFFu);                    // global_addr lo
    g0[3] = (unsigned)((ga >> 32) & 0x1FFFFFFu) | (2u << 30);// global_addr hi | type=2
    const unsigned td0 = 1u << 20, td1 = 1u << 20;           // huge tensor dims: no OOB clip
    v8u g1;
    g1[0] = (2u << 16) | (1u << 20) | (pint << 22) | (3u << 25); // 4B elems, pad_en, pad=4dw
    g1[1] = (td0 & 0xFFFFu) << 16;                           // tensor_dim0[15:0]
    g1[2] = (td0 >> 16) | ((td1 & 0xFFFFu) << 16);           // tensor_dim0[31:16]|tensor_dim1[15:0]
    g1[3] = (td1 >> 16) | (rowlen << 16);                    // tensor_dim1[31:16]|tile_dim0
    g1[4] = rows;                                            // tile_dim1 (tile_dim2=0)
    g1[5] = pitch_elems;                                     // tensor_dim0_stride[31:0]
    g1[6] = 0u;
    g1[7] = 0u;
    asm volatile("tensor_load_to_lds %0, %1" :: "s"(g0), "s"(g1) : "memory");
}

// ---------------- tiny kernels ----------------
__global__ void init_kernel(unsigned* bar) {
    if (threadIdx.x == 0) { bar[0] = 0u; }
}

// embs[t][:] = emb_table[ids[(t-1) mod T]][:]
__global__ void embs_kernel(const int* __restrict__ ids,
                            const float* __restrict__ table,
                            float* __restrict__ embs) {
    int t = blockIdx.x;
    int src = ids[(t + TT - 1) & (TT - 1)];
    const float4* row = (const float4*)(table + (size_t)src * EMB);
    float4* dst = (float4*)(embs + (size_t)t * EMB);
    for (int i = threadIdx.x; i < EMB / 4; i += blockDim.x) dst[i] = row[i];
}

// bias_x[n] = W_ih[n, E:] . [h0;c0] + b_ih[n] + b_hh[n]   (one wave per n)
__global__ void bias_kernel(const float* __restrict__ W_ih,
                            const float* __restrict__ b_ih,
                            const float* __restrict__ b_hh,
                            const float* __restrict__ h0,
                            const float* __restrict__ c0,
                            float* __restrict__ bias_x) {
    int wave = threadIdx.x >> 5, lane = threadIdx.x & 31;
    int n = blockIdx.x * 8 + wave;
    const float* wrow = W_ih + (size_t)n * WIH_LD + EMB;
    float s = 0.f;
    for (int i = lane; i < 2 * STATE; i += 32) {
        float f = (i < STATE) ? h0[i] : c0[i - STATE];
        s += wrow[i] * f;
    }
    for (int d = 16; d; d >>= 1) s += __shfl_down(s, d, 32);
    if (lane == 0) bias_x[n] = s + b_ih[n] + b_hh[n];
}

// zfix[v] = Z[v,:] . [h0;c0]   (one wave per v)
__global__ void zfix_kernel(const float* __restrict__ Z,
                            const float* __restrict__ h0,
                            const float* __restrict__ c0,
                            float* __restrict__ zfix) {
    int wave = threadIdx.x >> 5, lane = threadIdx.x & 31;
    int v = blockIdx.x * 8 + wave;
    const float* zr = Z + (size_t)v * (2 * STATE);
    float s = 0.f;
    for (int i = lane; i < 2 * STATE; i += 32) {
        float f = (i < STATE) ? h0[i] : c0[i - STATE];
        s += zr[i] * f;
    }
    for (int d = 16; d; d >>= 1) s += __shfl_down(s, d, 32);
    if (lane == 0) zfix[v] = s;
}

// ---------------- WMMA GEMM (transposed tiles, TDM-staged) ----------------
// C[t][v] = sum_k Tsrc[t][k] * Wsrc[v][k] + bias[v], computed as C^T tiles.
// Block tile: 128(v) x 64(t), Kc=32. 8 waves = 4(v) x 2(t), each wave a
// 32x32 subtile via 2x2 register blocking (4 v8f accumulators): every
// (a0,a1,b0,b1) fragment quad feeds 4 WMMAs -> 0.5 DS instr per WMMA.
#define RT 128    // v-rows per block
#define CT 64     // t-cols per block
#define TKC 32
#define LDSP (TKC + 4)
#define PINT 4u   // pad interval code: 32 DWORDs per row

__global__ void gemm_wmma_f32(const float* __restrict__ W1, int ldw1,
                              const float* __restrict__ W2, int ldw2,
                              const float* __restrict__ T1, int ldt1,
                              const float* __restrict__ T2, int ldt2,
                              int K1, int K2,
                              const float* __restrict__ bias,
                              float* __restrict__ C, int ldc) {
    __shared__ float Ws[2][RT][LDSP];
    __shared__ float Ts[2][CT][LDSP];

    const int r0 = blockIdx.x * RT;   // v / gate-row base
    const int t0 = blockIdx.y * CT;   // t base
    const int tid  = threadIdx.x;
    const int lane = tid & 31;
    const int wave = tid >> 5;
    const int wm = (wave & 3) * 32;   // v offset within block tile
    const int wn = (wave >> 2) * 32;  // t offset within block tile
    const int mA = lane & 15;
    const int koff = (lane >> 4) * 2;
    const bool issuer = (__builtin_amdgcn_readfirstlane((unsigned)wave) == 0u);

    const int Ktot = K1 + K2;
    const int np = Ktot / TKC;
    v8f acc00 = {}, acc01 = {}, acc10 = {}, acc11 = {};

    if (issuer) {  // panel 0
        tdm_load_tile((unsigned)(uintptr_t)&Ws[0][0][0], W1 + (size_t)r0 * ldw1, RT, TKC, ldw1, PINT);
        tdm_load_tile((unsigned)(uintptr_t)&Ts[0][0][0], T1 + (size_t)t0 * ldt1, CT, TKC, ldt1, PINT);
    }

    for (int p = 0; p < np; ++p) {
        if (issuer) {
            if (p + 1 < np) {   // prefetch next panel into other buffer
                int k0 = (p + 1) * TKC;
                const float* Wp; int lw; int kW;
                const float* Tp; int lt; int kT;
                if (k0 < K1) { Wp = W1; lw = ldw1; kW = k0;      Tp = T1; lt = ldt1; kT = k0; }
                else         { Wp = W2; lw = ldw2; kW = k0 - K1; Tp = T2; lt = ldt2; kT = k0 - K1; }
                int nb = (p + 1) & 1;
                tdm_load_tile((unsigned)(uintptr_t)&Ws[nb][0][0], Wp + (size_t)r0 * lw + kW, RT, TKC, lw, PINT);
                tdm_load_tile((unsigned)(uintptr_t)&Ts[nb][0][0], Tp + (size_t)t0 * lt + kT, CT, TKC, lt, PINT);
                __builtin_amdgcn_s_wait_tensorcnt((short)2);  // panel p done (in-order)
            } else {
                __builtin_amdgcn_s_wait_tensorcnt((short)0);
            }
        }
        __syncthreads();   // panel p visible to all waves

        const float (*Wb)[LDSP] = Ws[p & 1];
        const float (*Tb)[LDSP] = Ts[p & 1];
#pragma unroll
        for (int kk = 0; kk < TKC; kk += 4) {
            v2f a0, a1, b0, b1;
            a0[0] = Wb[wm + mA][kk + koff];
            a0[1] = Wb[wm + mA][kk + koff + 1];
            a1[0] = Wb[wm + 16 + mA][kk + koff];
            a1[1] = Wb[wm + 16 + mA][kk + koff + 1];
            b0[0] = Tb[wn + mA][kk + koff];
            b0[1] = Tb[wn + mA][kk + koff + 1];
            b1[0] = Tb[wn + 16 + mA][kk + koff];
            b1[1] = Tb[wn + 16 + mA][kk + koff + 1];
            acc00 = __builtin_amdgcn_wmma_f32_16x16x4_f32(false, a0, false, b0, (short)0, acc00, false, false);
            acc01 = __builtin_amdgcn_wmma_f32_16x16x4_f32(false, a0, false, b1, (short)0, acc01, false, false);
            acc10 = __builtin_amdgcn_wmma_f32_16x16x4_f32(false, a1, false, b0, (short)0, acc10, false, false);
            acc11 = __builtin_amdgcn_wmma_f32_16x16x4_f32(false, a1, false, b1, (short)0, acc11, false, false);
        }
        __syncthreads();   // all reads of buffer (p&1) done before TDM reuses it
    }

    // epilogue: each lane stores 8 consecutive v's per acc tile (b128 x2)
    const int hi8 = (lane >> 4) ? 8 : 0;
#pragma unroll
    for (int j = 0; j < 2; ++j) {
        const int tcol = t0 + wn + 16 * j + (lane & 15);
        float* orow = C + (size_t)tcol * ldc;
#pragma unroll
        for (int i = 0; i < 2; ++i) {
            const int vbase = r0 + wm + 16 * i + hi8;
            v8f acc = (j == 0) ? (i == 0 ? acc00 : acc10)
                               : (i == 0 ? acc01 : acc11);
            float4 c0 = *(const float4*)&bias[vbase];
            float4 c1 = *(const float4*)&bias[vbase + 4];
            float4 o0 = make_float4(acc[0] + c0.x, acc[1] + c0.y, acc[2] + c0.z, acc[3] + c0.w);
            float4 o1 = make_float4(acc[4] + c1.x, acc[5] + c1.y, acc[6] + c1.z, acc[7] + c1.w);
            *(float4*)(orow + vbase)     = o0;
            *(float4*)(orow + vbase + 4) = o1;
        }
    }
}

// ---------------- persistent LSTM scan ----------------
#define JPB 8
__global__ void scan_kernel(const float* __restrict__ gates_x,
                            const float* __restrict__ W_hh,
                            const float* __restrict__ h0,
                            const float* __restrict__ c0,
                            float* __restrict__ hs,
                            unsigned* __restrict__ bar) {
    __shared__ float h_s[STATE];
    __shared__ float gs[32];
    __shared__ float c_s[JPB];

    const int tid = threadIdx.x;
    const int b = blockIdx.x;
    const int j0 = b * JPB;
    if (tid < JPB) c_s[tid] = c0[j0 + tid];

    const int d = tid >> 3;          // dot id 0..31
    const int part = tid & 7;        // 8 lanes per dot
    const int g = d >> 3;            // gate 0..3
    const int jj = d & 7;
    const int n = g * STATE + j0 + jj;
    const float4* wrow = (const float4*)(W_hh + (size_t)n * STATE) + part * 32;

    for (int t = 0; t < TT; ++t) {
        const float* h_prev = (t == 0) ? h0 : (hs + (size_t)(t - 1) * STATE);
#pragma unroll
        for (int q = 0; q < 4; ++q) {
            int idx = tid + q * 256;
            h_s[idx] = __hip_atomic_load((const float*)&h_prev[idx],
                                         __ATOMIC_RELAXED, __HIP_MEMORY_SCOPE_AGENT);
        }
        __syncthreads();

        float s = 0.f;
        const float4* hp4 = (const float4*)&h_s[part * 128];
#pragma unroll 8
        for (int q = 0; q < 32; ++q) {
            float4 w = wrow[q];
            float4 h4 = hp4[q];
            s += w.x * h4.x + w.y * h4.y + w.z * h4.z + w.w * h4.w;
        }
        s += __shfl_down(s, 4, 8);
        s += __shfl_down(s, 2, 8);
        s += __shfl_down(s, 1, 8);
        if (part == 0) gs[d] = s + gates_x[(size_t)t * G4 + n];
        __syncthreads();

        if (tid < JPB) {
            float iv = sigmoidf_(gs[tid]);
            float fv = sigmoidf_(gs[8 + tid]);
            float gv = tanhf(gs[16 + tid]);
            float ov = sigmoidf_(gs[24 + tid]);
            float c = fv * c_s[tid] + iv * gv;
            c_s[tid] = c;
            float h = ov * tanhf(c);
            __hip_atomic_store(&hs[(size_t)t * STATE + j0 + tid], h,
                               __ATOMIC_RELAXED, __HIP_MEMORY_SCOPE_AGENT);
        }
        grid_barrier(bar, bar + 1, SCAN_BLOCKS);
    }
}

// ---------------- launch ----------------
extern "C" void kernel_launch(void* const* d_in, const int* in_sizes, int n_in,
                              void* d_out, int out_size, void* d_ws, size_t ws_size,
                              hipStream_t stream) {
    const float* h0        = (const float*)d_in[0];
    const float* c0        = (const float*)d_in[1];
    const int*   ids       = (const int*)  d_in[2];
    const float* emb_table = (const float*)d_in[3];
    const float* W_ih      = (const float*)d_in[4];
    const float* W_hh      = (const float*)d_in[5];
    const float* b_ih      = (const float*)d_in[6];
    const float* b_hh      = (const float*)d_in[7];
    const float* X         = (const float*)d_in[8];
    const float* Y         = (const float*)d_in[9];
    const float* Z         = (const float*)d_in[10];
    float* out = (float*)d_out;

    float* ws      = (float*)d_ws;
    float* embs    = ws;                           // 512*512
    float* bias_x  = embs + (size_t)TT * EMB;      // 4096
    float* zfix    = bias_x + G4;                  // 32000
    float* gates_x = zfix + VOC;                   // 512*4096
    float* hs      = gates_x + (size_t)TT * G4;    // 512*1024
    unsigned* bar  = (unsigned*)(hs + (size_t)TT * STATE);  // 2 uints

    init_kernel<<<1, 32, 0, stream>>>(bar);
    embs_kernel<<<TT, 128, 0, stream>>>(ids, emb_table, embs);
    bias_kernel<<<G4 / 8, 256, 0, stream>>>(W_ih, b_ih, b_hh, h0, c0, bias_x);
    zfix_kernel<<<VOC / 8, 256, 0, stream>>>(Z, h0, c0, zfix);

    // gates_x[T x 4096] = embs @ W_ih[:, :E]^T + bias_x
    gemm_wmma_f32<<<dim3(G4 / RT, TT / CT), 256, 0, stream>>>(
        W_ih, WIH_LD, nullptr, 0,
        embs, EMB, nullptr, 0,
        EMB, 0,
        bias_x, gates_x, G4);

    scan_kernel<<<SCAN_BLOCKS, 256, 0, stream>>>(gates_x, W_hh, h0, c0, hs, bar);

    // scores[T x VOC] = [hs | embs] @ [X | Y]^T + zfix
    gemm_wmma_f32<<<dim3(VOC / RT, TT / CT), 256, 0, stream>>>(
        X, STATE, Y, EMB,
        hs, STATE, embs, EMB,
        STATE, EMB,
        zfix, out, VOC);
}